// EdgeDense_52080773431962
// MI455X (gfx1250) — compile-verified
//
#include <hip/hip_runtime.h>

typedef __attribute__((ext_vector_type(2))) float v2f;
typedef __attribute__((ext_vector_type(4))) float v4f;
typedef __attribute__((ext_vector_type(8))) float v8f;

#define IN_F  128
#define OUT_F 32

// ---------------------------------------------------------------------------
// Phase 1:  z = x @ W + b   via V_WMMA_F32_16X16X4_F32  (full f32 precision)
//
// One wave computes one 16x16 tile of z.  M-tiles = n_nodes/16, N-tiles = 2.
// A (16x4 f32) layout per ISA 7.12.2:  lane l in [0,15], half h = lane>>4:
//   VGPR0 = A[l][2h], VGPR1 = A[l][2h+1]          -> one float2 load per step
// B (4x16 f32) mirrored:  VGPR0 = W[k+2h][n0+l], VGPR1 = W[k+2h+1][n0+l]
// C/D (16x16 f32, 8 VGPRs): c[v] = row m0 + v + 8h, col n0 + l
// ---------------------------------------------------------------------------
__global__ __launch_bounds__(256) void edense_z_wmma(
    const float* __restrict__ x, const float* __restrict__ W,
    const float* __restrict__ b, float* __restrict__ z, int n_nodes)
{
    const int mtiles = (n_nodes + 15) >> 4;
    const int ntiles = OUT_F / 16;                       // == 2
    const int wave   = blockIdx.x * (blockDim.x >> 5) + (threadIdx.x >> 5);
    if (wave >= mtiles * ntiles) return;                 // wave-uniform exit

    const int mt   = wave >> 1;
    const int nt   = wave & 1;
    const int m0   = mt << 4;
    const int n0   = nt << 4;
    const int lane = threadIdx.x & 31;
    const int half = lane >> 4;                          // selects K pair
    const int l    = lane & 15;

    // Clamp row for the A loads so EXEC stays all-ones for WMMA; the store
    // below guards the actual writes. (n_nodes=100000 is a multiple of 16,
    // so this is purely defensive.)
    int arow = m0 + l;
    if (arow >= n_nodes) arow = n_nodes - 1;
    const float* xrow = x + (size_t)arow * IN_F + 2 * half;
    const float* wcol = W + (size_t)(2 * half) * OUT_F + n0 + l;

    // Accumulator seeded with bias: every row of the tile gets b[n0+l].
    const float bv = b[n0 + l];
    v8f c;
#pragma unroll
    for (int v = 0; v < 8; ++v) c[v] = bv;

#pragma unroll
    for (int k = 0; k < IN_F; k += 4) {
        v2f a = *(const v2f*)(xrow + k);                 // A[l][k+2h .. k+2h+1]
        v2f bb;
        bb[0] = wcol[(size_t)k * OUT_F];                 // W[k+2h  ][n0+l]
        bb[1] = wcol[(size_t)k * OUT_F + OUT_F];         // W[k+2h+1][n0+l]
        c = __builtin_amdgcn_wmma_f32_16x16x4_f32(
                /*neg_a=*/false, a, /*neg_b=*/false, bb,
                /*c_mod=*/(short)0, c, /*reuse_a=*/false, /*reuse_b=*/false);
    }

#pragma unroll
    for (int v = 0; v < 8; ++v) {
        const int row = m0 + v + 8 * half;
        if (row < n_nodes)
            z[(size_t)row * OUT_F + n0 + l] = c[v];
    }
}

// ---------------------------------------------------------------------------
// Phase 2:  out[e] = vals[e] * (z[rows[e]] + z[cols[e]])
//
// Pure bandwidth (z gathers hit L2; output store dominates HBM traffic).
// 8 lanes per edge, one float4 eighth-row each: every wave-level store
// instruction covers 4 edges x 128B = 512B fully contiguous.
// ---------------------------------------------------------------------------
__global__ __launch_bounds__(256) void edense_edge_kernel(
    const float* __restrict__ z, const int* __restrict__ rows,
    const int* __restrict__ cols, const float* __restrict__ vals,
    float* __restrict__ out, int n_edges)
{
    const long long gid = (long long)blockIdx.x * blockDim.x + threadIdx.x;
    if (gid >= (long long)n_edges * 8) return;
    const int e = (int)(gid >> 3);
    const int q = ((int)gid & 7) << 2;                   // column offset 0..28

    const int   r = rows[e];
    const int   cI = cols[e];
    const float w = vals[e];

    const v4f zr = *(const v4f*)(z + (size_t)r  * OUT_F + q);
    const v4f zc = *(const v4f*)(z + (size_t)cI * OUT_F + q);
    v4f o = w * (zr + zc);
    *(v4f*)(out + (size_t)e * OUT_F + q) = o;
}

// ---------------------------------------------------------------------------
// Launch: d_in order = {x, W, b, edge_rows, edge_cols, edge_vals}
// d_ws holds z (n_nodes * 32 floats = 12.8 MB).
// ---------------------------------------------------------------------------
extern "C" void kernel_launch(void* const* d_in, const int* in_sizes, int n_in,
                              void* d_out, int out_size, void* d_ws, size_t ws_size,
                              hipStream_t stream) {
    const float* x    = (const float*)d_in[0];
    const float* W    = (const float*)d_in[1];
    const float* b    = (const float*)d_in[2];
    const int*   rows = (const int*)d_in[3];
    const int*   cols = (const int*)d_in[4];
    const float* vals = (const float*)d_in[5];
    float*       out  = (float*)d_out;
    float*       z    = (float*)d_ws;

    const int n_nodes = in_sizes[0] / IN_F;
    const int n_edges = in_sizes[5];

    // Phase 1: one wave per 16x16 tile, 8 waves per 256-thread block.
    const int mtiles = (n_nodes + 15) / 16;
    const int tiles  = mtiles * (OUT_F / 16);
    const int blocks1 = (tiles + 7) / 8;
    edense_z_wmma<<<blocks1, 256, 0, stream>>>(x, W, b, z, n_nodes);

    // Phase 2: 8 threads per edge (stream-ordered after phase 1).
    const long long threads2 = (long long)n_edges * 8;
    const int blocks2 = (int)((threads2 + 255) / 256);
    edense_edge_kernel<<<blocks2, 256, 0, stream>>>(z, rows, cols, vals, out, n_edges);
}